// MinMaxLayer_29755533427373
// MI455X (gfx1250) — compile-verified
//
#include <hip/hip_runtime.h>

#define R_OUT 5
#define T_LEN 8192
#define BLOCK 256
#define NWAVES (BLOCK / 32)
#define STAGE_ELEMS (BLOCK * 4)          // 1024 floats = 4 KB per stage
#define NSTAGES (T_LEN / STAGE_ELEMS)    // 8
#define DEPTH 4                          // async pipeline depth

#if defined(__AMDGCN__) && defined(__gfx1250__) && \
    __has_builtin(__builtin_amdgcn_global_load_async_to_lds_b128) && \
    __has_builtin(__builtin_amdgcn_s_wait_asynccnt)
#define USE_ASYNC_LDS 1
#else
#define USE_ASYNC_LDS 0
#endif

#if USE_ASYNC_LDS
typedef int v4i __attribute__((vector_size(16)));
typedef __attribute__((address_space(1))) v4i* gv4i_ptr;
typedef __attribute__((address_space(3))) v4i* lv4i_ptr;

__device__ __forceinline__ void wait_async_upto(int n) {
    // s_wait_asynccnt requires a literal immediate; dispatch on constants.
    switch (n) {
        case 0:  __builtin_amdgcn_s_wait_asynccnt(0); break;
        case 1:  __builtin_amdgcn_s_wait_asynccnt(1); break;
        case 2:  __builtin_amdgcn_s_wait_asynccnt(2); break;
        default: __builtin_amdgcn_s_wait_asynccnt(3); break;
    }
}
#endif

// Accumulate one 4-element group.
// Fast path: group lies entirely in the *exclusive* region of one window
// [ex_lo[r], ex_hi[r]) = [end[r-1], start[r+1]) -> fold group min/max once.
// Slow path (boundary/overlap/L-edge straddlers, ~0.5% of groups): exact
// per-element masked accumulate against [start[r], end[r]).
__device__ __forceinline__ void accum_group(float4 v, int t0,
                                            const int* wstart, const int* wend,
                                            const int* ex_lo, const int* ex_hi,
                                            float* mx, float* mn) {
    const float gmx = fmaxf(fmaxf(v.x, v.y), fmaxf(v.z, v.w));
    const float gmn = fminf(fminf(v.x, v.y), fminf(v.z, v.w));
    bool anyfull = false;
#pragma unroll
    for (int r = 0; r < R_OUT; ++r) {
        const bool full = (t0 >= ex_lo[r]) && (t0 + 3 < ex_hi[r]);
        mx[r] = fmaxf(mx[r], full ? gmx : -__builtin_inff());
        mn[r] = fminf(mn[r], full ? gmn :  __builtin_inff());
        anyfull |= full;
    }
    if (!anyfull && t0 < ex_hi[R_OUT - 1]) {   // ex_hi[4] == L
        const float vv[4] = {v.x, v.y, v.z, v.w};
#pragma unroll
        for (int j = 0; j < 4; ++j) {
            const int t = t0 + j;
            const float x = vv[j];
#pragma unroll
            for (int r = 0; r < R_OUT; ++r) {
                const bool in = (t >= wstart[r]) && (t < wend[r]);
                mx[r] = fmaxf(mx[r], in ? x : -__builtin_inff());
                mn[r] = fminf(mn[r], in ? x :  __builtin_inff());
            }
        }
    }
}

__global__ __launch_bounds__(BLOCK) void minmax_kernel(
    const float* __restrict__ inputs,
    const int*   __restrict__ lengths,
    float*       __restrict__ out)
{
    const int b   = blockIdx.x;
    const int tid = threadIdx.x;
    const float* row = inputs + (size_t)b * T_LEN;
    const int L = lengths[b];          // workgroup-uniform -> scalar regs

    int wstart[R_OUT], wend[R_OUT], ex_lo[R_OUT], ex_hi[R_OUT];
#pragma unroll
    for (int r = 0; r < R_OUT; ++r) {
        wstart[r] = (r * L) / R_OUT;
        wend[r]   = ((r + 1) * L + (R_OUT - 1)) / R_OUT;
    }
#pragma unroll
    for (int r = 0; r < R_OUT; ++r) {
        ex_lo[r] = (r == 0) ? 0 : wend[r - 1];             // first t exclusive to r
        ex_hi[r] = (r == R_OUT - 1) ? wend[R_OUT - 1]      // == L
                                    : wstart[r + 1];       // first t shared with r+1
    }

    float mx[R_OUT], mn[R_OUT];
#pragma unroll
    for (int r = 0; r < R_OUT; ++r) { mx[r] = -__builtin_inff(); mn[r] = __builtin_inff(); }

#if USE_ASYNC_LDS
    // ---- CDNA5 async global->LDS streaming pipeline (ASYNCcnt-tracked) ----
    __shared__ __align__(16) float stage[DEPTH][STAGE_ELEMS];

    // prologue: fill the pipeline with DEPTH-1 stages
#pragma unroll
    for (int k = 0; k < DEPTH - 1; ++k) {
        gv4i_ptr gsrc = (gv4i_ptr)(row + k * STAGE_ELEMS + tid * 4);
        lv4i_ptr ldst = (lv4i_ptr)(&stage[k][tid * 4]);
        __builtin_amdgcn_global_load_async_to_lds_b128(gsrc, ldst, 0, 0);
    }

    for (int s = 0; s < NSTAGES; ++s) {
        if (s + DEPTH - 1 < NSTAGES) {
            gv4i_ptr gsrc = (gv4i_ptr)(row + (s + DEPTH - 1) * STAGE_ELEMS + tid * 4);
            lv4i_ptr ldst = (lv4i_ptr)(&stage[(s + DEPTH - 1) % DEPTH][tid * 4]);
            __builtin_amdgcn_global_load_async_to_lds_b128(gsrc, ldst, 0, 0);
            __builtin_amdgcn_s_wait_asynccnt(DEPTH - 1);   // oldest stage retired
        } else {
            wait_async_upto(NSTAGES - 1 - s);              // drain tail
        }
        asm volatile("" ::: "memory");   // LDS now holds stage s for this wave's lanes
        const float4 v = *(const float4*)&stage[s % DEPTH][tid * 4];
        accum_group(v, s * STAGE_ELEMS + tid * 4, wstart, wend, ex_lo, ex_hi, mx, mn);
    }
#else
    // ---- fallback: direct wide vector loads ----
    for (int s = 0; s < NSTAGES; ++s) {
        const float4 v = *(const float4*)(row + s * STAGE_ELEMS + tid * 4);
        accum_group(v, s * STAGE_ELEMS + tid * 4, wstart, wend, ex_lo, ex_hi, mx, mn);
    }
#endif

    // ---- wave32 reduction ----
#pragma unroll
    for (int off = 16; off >= 1; off >>= 1) {
#pragma unroll
        for (int r = 0; r < R_OUT; ++r) {
            mx[r] = fmaxf(mx[r], __shfl_xor(mx[r], off, 32));
            mn[r] = fminf(mn[r], __shfl_xor(mn[r], off, 32));
        }
    }

    // ---- cross-wave reduction via LDS ----
    __shared__ float pmax[NWAVES][R_OUT];
    __shared__ float pmin[NWAVES][R_OUT];
    const int wave = tid >> 5;
    const int lane = tid & 31;
    if (lane == 0) {
#pragma unroll
        for (int r = 0; r < R_OUT; ++r) { pmax[wave][r] = mx[r]; pmin[wave][r] = mn[r]; }
    }
    __syncthreads();

    __shared__ float vals[2 * R_OUT];
    if (tid < 2 * R_OUT) {
        float v;
        if (tid < R_OUT) {
            v = -__builtin_inff();
#pragma unroll
            for (int w = 0; w < NWAVES; ++w) v = fmaxf(v, pmax[w][tid]);
        } else {
            v = __builtin_inff();
#pragma unroll
            for (int w = 0; w < NWAVES; ++w) v = fminf(v, pmin[w][tid - R_OUT]);
        }
        vals[tid] = v;
    }
    __syncthreads();

    // ---- stable rank sort of 10 values, 10 threads in parallel ----
    if (tid < 2 * R_OUT) {
        const float v = vals[tid];
        int rank = 0;
#pragma unroll
        for (int j = 0; j < 2 * R_OUT; ++j) {
            const float u = vals[j];
            rank += (u < v) || (u == v && j < tid);
        }
        out[(size_t)b * (2 * R_OUT) + rank] = v;
    }
}

extern "C" void kernel_launch(void* const* d_in, const int* in_sizes, int n_in,
                              void* d_out, int out_size, void* d_ws, size_t ws_size,
                              hipStream_t stream) {
    const float* inputs  = (const float*)d_in[0];   // (B, T) f32
    const int*   lengths = (const int*)d_in[1];     // (B,)  i32
    float*       out     = (float*)d_out;           // (B, 10) f32
    const int B = in_sizes[1];                      // 2048
    minmax_kernel<<<dim3(B), dim3(BLOCK), 0, stream>>>(inputs, lengths, out);
}